// qcir_35854386987509
// MI455X (gfx1250) — compile-verified
//
#include <hip/hip_runtime.h>
#include <math.h>

#define NQ    12
#define DIM   4096      // 2^12
#define BATCH 256

typedef float v2f __attribute__((ext_vector_type(2)));
typedef float v8f __attribute__((ext_vector_type(8)));

struct Shared {
    float psi_re[DIM];
    float psi_im[DIM];
    unsigned short perm1[DIM];
    unsigned short perm2[DIM];
    float U_re[16 * 17];     // padded rows -> no LDS bank conflict on A loads
    float U_im[16 * 17];
    float g_re[NQ][4];       // fused 2x2 gate per qubit, row*2+col
    float g_im[NQ][4];
    float red[8 * NQ];       // cross-wave reduction
    float enc_c[NQ];
    float enc_s[NQ];
};

// ring-of-CNOT net index map: sequential "if control bit -> toggle target bit"
__device__ inline unsigned short ring_perm(int k, int step) {
    int kk = k;
#pragma unroll
    for (int i = 0; i < NQ; ++i) {
        int cpos = 11 - i;
        int tpos = 11 - ((i + step) % NQ);
        if ((kk >> cpos) & 1) kk ^= (1 << tpos);
    }
    return (unsigned short)kk;
}

// address of element (row b in group G, column c) in the 16x256 reshape
template <int G>
__device__ inline int gaddr(int row, int col) {
    if (G == 0)      return (row << 8) + col;                               // qubits 0-3 (bits 11..8)
    else if (G == 1) return ((col >> 4) << 8) + (row << 4) + (col & 15);    // qubits 4-7 (bits 7..4)
    else             return (col << 4) + row;                               // qubits 8-11 (bits 3..0)
}

// fused Ry(c)*Rz(b)*Ry(a) per qubit
__device__ inline void compute_yzy(Shared& sh, const float* __restrict__ w, int off) {
    int q = threadIdx.x;
    if (q < NQ) {
        float a = w[off + 3 * q + 0] * 0.5f;
        float b = w[off + 3 * q + 1] * 0.5f;
        float c = w[off + 3 * q + 2] * 0.5f;
        float ca = cosf(a), sa = sinf(a);
        float cb = cosf(b), sb = sinf(b);
        float cc = cosf(c), sc = sinf(c);
        float cpc = cc * ca - sc * sa;   // cos((a+c)/2)
        float cmc = cc * ca + sc * sa;   // cos((a-c)/2)
        float sps = sa * cc + ca * sc;   // sin((a+c)/2)
        float sms = cc * sa - sc * ca;   // sin((a-c)/2)
        sh.g_re[q][0] =  cb * cpc;  sh.g_im[q][0] = -sb * cmc;
        sh.g_re[q][1] = -cb * sps;  sh.g_im[q][1] =  sb * sms;
        sh.g_re[q][2] =  cb * sps;  sh.g_im[q][2] =  sb * sms;
        sh.g_re[q][3] =  cb * cpc;  sh.g_im[q][3] =  sb * cmc;
    }
}

__device__ inline void compute_ry(Shared& sh, const float* __restrict__ w, int off) {
    int q = threadIdx.x;
    if (q < NQ) {
        float h = w[off + q] * 0.5f;
        float c = cosf(h), s = sinf(h);
        sh.g_re[q][0] = c;  sh.g_re[q][1] = -s;
        sh.g_re[q][2] = s;  sh.g_re[q][3] = c;
        sh.g_im[q][0] = 0.f; sh.g_im[q][1] = 0.f;
        sh.g_im[q][2] = 0.f; sh.g_im[q][3] = 0.f;
    }
}

// 16x16 group unitary = kron of 4 fused 2x2 gates (first qubit of group = MSB)
__device__ inline void compute_U(Shared& sh, int g) {
    int e = threadIdx.x;
    int a = e >> 4, b = e & 15;
    int q0 = 4 * g;
    int idx = ((((a >> 3) & 1) << 1) | ((b >> 3) & 1));
    float ur = sh.g_re[q0][idx], ui = sh.g_im[q0][idx];
#pragma unroll
    for (int j = 1; j < 4; ++j) {
        int bit = 3 - j;
        int id2 = ((((a >> bit) & 1) << 1) | ((b >> bit) & 1));
        float gr = sh.g_re[q0 + j][id2], gi = sh.g_im[q0 + j][id2];
        float nr = ur * gr - ui * gi;
        float ni = ur * gi + ui * gr;
        ur = nr; ui = ni;
    }
    sh.U_re[a * 17 + b] = ur;
    sh.U_im[a * 17 + b] = ui;
}

// D[16x256] = U * P, complex, in-place on psi.  K=16 via 4 chained f32 WMMAs.
template <int G>
__device__ inline void apply_group(Shared& sh, bool cplx) {
    const int lane = threadIdx.x & 31;
    const int wave = threadIdx.x >> 5;
    const int l16  = lane & 15;
    const int hi   = lane >> 4;

    // A fragments (shared by both tiles of this wave)
    v2f Ar[4], Ai[4], An[4];
#pragma unroll
    for (int kk = 0; kk < 4; ++kk) {
        int b0 = kk * 4 + hi * 2;
        Ar[kk][0] = sh.U_re[l16 * 17 + b0];
        Ar[kk][1] = sh.U_re[l16 * 17 + b0 + 1];
        Ai[kk][0] = sh.U_im[l16 * 17 + b0];
        Ai[kk][1] = sh.U_im[l16 * 17 + b0 + 1];
        An[kk][0] = -Ai[kk][0];
        An[kk][1] = -Ai[kk][1];
    }

#pragma unroll
    for (int t = 0; t < 2; ++t) {
        int col = (wave * 2 + t) * 16 + l16;
        v2f Br[4], Bi[4];
#pragma unroll
        for (int kk = 0; kk < 4; ++kk) {
            int r0 = kk * 4 + hi * 2;
            Br[kk][0] = sh.psi_re[gaddr<G>(r0,     col)];
            Br[kk][1] = sh.psi_re[gaddr<G>(r0 + 1, col)];
            Bi[kk][0] = sh.psi_im[gaddr<G>(r0,     col)];
            Bi[kk][1] = sh.psi_im[gaddr<G>(r0 + 1, col)];
        }
        v8f dre = {}, dim_ = {};
#pragma unroll
        for (int kk = 0; kk < 4; ++kk)
            dre = __builtin_amdgcn_wmma_f32_16x16x4_f32(false, Ar[kk], false, Br[kk],
                                                        (short)0, dre, false, false);
#pragma unroll
        for (int kk = 0; kk < 4; ++kk)
            dim_ = __builtin_amdgcn_wmma_f32_16x16x4_f32(false, Ar[kk], false, Bi[kk],
                                                         (short)0, dim_, false, false);
        if (cplx) {
#pragma unroll
            for (int kk = 0; kk < 4; ++kk)
                dre = __builtin_amdgcn_wmma_f32_16x16x4_f32(false, An[kk], false, Bi[kk],
                                                            (short)0, dre, false, false);
#pragma unroll
            for (int kk = 0; kk < 4; ++kk)
                dim_ = __builtin_amdgcn_wmma_f32_16x16x4_f32(false, Ai[kk], false, Br[kk],
                                                             (short)0, dim_, false, false);
        }
#pragma unroll
        for (int v = 0; v < 8; ++v) {
            int a = gaddr<G>(v + hi * 8, col);
            sh.psi_re[a] = dre[v];
            sh.psi_im[a] = dim_[v];
        }
    }
}

template <int G>
__device__ inline void group_pass(Shared& sh, bool cplx) {
    compute_U(sh, G);
    __syncthreads();
    apply_group<G>(sh, cplx);
    __syncthreads();
}

__device__ inline void gate_layers(Shared& sh, bool cplx) {
    group_pass<0>(sh, cplx);
    group_pass<1>(sh, cplx);
    group_pass<2>(sh, cplx);
}

__device__ inline void apply_perm(Shared& sh, const unsigned short* perm) {
    int t = threadIdx.x;
    float r[16], m[16];
#pragma unroll
    for (int j = 0; j < 16; ++j) {
        int k = t + j * 256;
        r[j] = sh.psi_re[k];
        m[j] = sh.psi_im[k];
    }
    __syncthreads();
#pragma unroll
    for (int j = 0; j < 16; ++j) {
        int k = t + j * 256;
        int d = perm[k];
        sh.psi_re[d] = r[j];
        sh.psi_im[d] = m[j];
    }
    __syncthreads();
}

__global__ __launch_bounds__(256) void qsim_kernel(const float* __restrict__ x,
                                                   const float* __restrict__ w,
                                                   float* __restrict__ out) {
    __shared__ Shared sh;
    const int sample = blockIdx.x;
    const int tid = threadIdx.x;

    // precompute CNOT-ring permutations (reused 5x / 2x)
#pragma unroll
    for (int j = 0; j < 16; ++j) {
        int k = tid + j * 256;
        sh.perm1[k] = ring_perm(k, 1);
        sh.perm2[k] = ring_perm(k, 2);
    }
    // encoding angles
    if (tid < NQ) {
        float h = x[sample * NQ + tid] * 0.5f;
        sh.enc_c[tid] = cosf(h);
        sh.enc_s[tid] = sinf(h);
    }
    __syncthreads();

    // |0..0> through the Ry encoding layer is a product state: build directly
#pragma unroll
    for (int j = 0; j < 16; ++j) {
        int k = tid + j * 256;
        float p = 1.f;
#pragma unroll
        for (int i = 0; i < NQ; ++i)
            p *= ((k >> (11 - i)) & 1) ? sh.enc_s[i] : sh.enc_c[i];
        sh.psi_re[k] = p;
        sh.psi_im[k] = 0.f;
    }
    __syncthreads();

    compute_yzy(sh, w, 0);   __syncthreads(); gate_layers(sh, true); apply_perm(sh, sh.perm1);
    compute_yzy(sh, w, 36);  __syncthreads(); gate_layers(sh, true); apply_perm(sh, sh.perm1);
    compute_yzy(sh, w, 72);  __syncthreads(); gate_layers(sh, true); apply_perm(sh, sh.perm2);
    compute_yzy(sh, w, 108); __syncthreads(); gate_layers(sh, true); apply_perm(sh, sh.perm2);
#pragma unroll
    for (int r = 0; r < 3; ++r) {            // BasicEntangler: Ry layer + ring(1)
        compute_ry(sh, w, 144 + r * NQ);
        __syncthreads();
        gate_layers(sh, false);
        apply_perm(sh, sh.perm1);
    }
    compute_yzy(sh, w, 180); __syncthreads(); gate_layers(sh, true);

    // <Z_i> = 4 * sum_k |psi_k|^2 * (1 - 2*bit_i(k))
    float acc[NQ];
#pragma unroll
    for (int i = 0; i < NQ; ++i) acc[i] = 0.f;
#pragma unroll
    for (int j = 0; j < 16; ++j) {
        int k = tid + j * 256;
        float re = sh.psi_re[k], im = sh.psi_im[k];
        float p = re * re + im * im;
#pragma unroll
        for (int i = 0; i < NQ; ++i)
            acc[i] += ((k >> (11 - i)) & 1) ? -p : p;
    }
#pragma unroll
    for (int off = 16; off >= 1; off >>= 1)
#pragma unroll
        for (int i = 0; i < NQ; ++i)
            acc[i] += __shfl_xor(acc[i], off, 32);

    const int lane = tid & 31, wave = tid >> 5;
    if (lane == 0)
#pragma unroll
        for (int i = 0; i < NQ; ++i) sh.red[wave * NQ + i] = acc[i];
    __syncthreads();
    if (tid < NQ) {
        float s = 0.f;
#pragma unroll
        for (int wv = 0; wv < 8; ++wv) s += sh.red[wv * NQ + tid];
        out[sample * NQ + tid] = 4.0f * s;
    }
}

extern "C" void kernel_launch(void* const* d_in, const int* in_sizes, int n_in,
                              void* d_out, int out_size, void* d_ws, size_t ws_size,
                              hipStream_t stream) {
    const float* x = (const float*)d_in[0];   // (256, 12) float32
    const float* w = (const float*)d_in[1];   // (216,)    float32
    float* out = (float*)d_out;               // (256, 12) float32
    qsim_kernel<<<dim3(BATCH), dim3(256), 0, stream>>>(x, w, out);
}